// BSplineActivation_25812753449581
// MI455X (gfx1250) — compile-verified
//
#include <hip/hip_runtime.h>
#include <hip/hip_bf16.h>

// ---------------------------------------------------------------------------
// Cubic B-spline activation, 1024 uniform knots on [-pi, pi], 1020 weights.
// B is 4-banded -> O(1) per point. Memory bound: ~2MB total traffic.
// Weights staged into LDS via the CDNA5 Tensor Data Mover (tensor_load_to_lds),
// tracked with TENSORcnt.
// ---------------------------------------------------------------------------

#define NW        1020      // NUM_KNOTS - DEGREE - 1
#define NSEG      1023      // NUM_KNOTS - 1 intervals
#define BLOCK     256

typedef unsigned int u32x4 __attribute__((ext_vector_type(4)));
typedef int          i32x4 __attribute__((ext_vector_type(4)));
typedef int          i32x8 __attribute__((ext_vector_type(8)));

// Stage NW floats from global into LDS. Wave 0 issues one TDM descriptor
// (TDM ignores EXEC, so only one wave may execute the instruction), waits on
// TENSORcnt, then the workgroup barrier publishes LDS to all waves.
__device__ __forceinline__ void stage_weights(const float* __restrict__ wsrc,
                                              float* __restrict__ lds_dst) {
#if defined(__has_builtin) && __has_builtin(__builtin_amdgcn_tensor_load_to_lds) && \
    __has_builtin(__builtin_amdgcn_s_wait_tensorcnt)
  if (threadIdx.x < 32) {  // wave 0 only (uniform branch)
    unsigned long long gaddr = (unsigned long long)(size_t)wsrc;
    // Generic LDS pointer = {shared aperture, lds_offset}; low 32 bits are the
    // byte offset the DS/TDM hardware consumes.
    unsigned int laddr = (unsigned int)((unsigned long long)(size_t)lds_dst & 0xFFFFFFFFull);

    // ---- D# group 0 (128b): count=1 (valid), lds_addr, global_addr, type=2
    u32x4 g0;
    g0[0] = 1u;                                                  // count=1, user mode, no gather
    g0[1] = laddr;                                               // lds_addr[63:32 of g0]
    g0[2] = (unsigned int)(gaddr & 0xFFFFFFFFull);               // global_addr[31:0]
    g0[3] = (unsigned int)((gaddr >> 32) & 0x1FFFFFFull)         // global_addr[56:32]
          | (2u << 30);                                          // type = 2 ("image")

    // ---- D# group 1 (256b): data_size=4B, 1 row of NW elements
    i32x8 g1;
    g1[0] = (2 << 16);                         // wg_mask=0 (not in cluster), data_size=2 (4B)
    g1[1] = (int)(((unsigned)NW & 0xFFFFu) << 16);   // tensor_dim0[15:0] @ bits[63:48]
    g1[2] = (int)((((unsigned)NW >> 16) & 0xFFFFu)   // tensor_dim0[31:16]
          | (1u << 16));                             // tensor_dim1[15:0] = 1
    g1[3] = (int)((unsigned)NW << 16);               // tensor_dim1 hi = 0, tile_dim0 = NW
    g1[4] = 1;                                       // tile_dim1 = 1, tile_dim2 = 0 (unused)
    g1[5] = NW;                                      // tensor_dim0_stride[31:0]
    g1[6] = 0;                                       // stride hi / tensor_dim1_stride lo
    g1[7] = 0;                                       // tensor_dim1_stride hi

    i32x4 gz = {0, 0, 0, 0};                         // groups 2/3: dims 2..4 unused

#if __has_include(<hip/amd_detail/amd_gfx1250_TDM.h>)
    // therock / clang-23 lane: 6-arg builtin
    i32x8 gz8 = {0, 0, 0, 0, 0, 0, 0, 0};
    __builtin_amdgcn_tensor_load_to_lds(g0, g1, gz, gz, gz8, 0);
#else
    // ROCm 7.2 / clang-22 lane: 5-arg builtin
    __builtin_amdgcn_tensor_load_to_lds(g0, g1, gz, gz, 0);
#endif
    __builtin_amdgcn_s_wait_tensorcnt(0);            // s_wait_tensorcnt 0
  }
  __syncthreads();
#else
  // Fallback (also what the host pass parses): cooperative copy.
  for (int j = threadIdx.x; j < NW; j += blockDim.x) lds_dst[j] = wsrc[j];
  __syncthreads();
#endif
}

// Evaluate spline at one point using LDS-resident weights.
__device__ __forceinline__ float bspline_eval(float xf, const float* __restrict__ lw) {
  const double PI_D = 3.14159265358979323846;
  const double INVH = (double)NSEG / (2.0 * PI_D);       // 1023 / (2*pi)
  double sd = ((double)xf + PI_D) * INVH;                // knot-space coordinate
  // Outside [t0, t_n) (or NaN): every level-0 seed is zero -> output 0.
  if (!(sd >= 0.0) || sd >= (double)NSEG) return 0.0f;
  int m = (int)sd;                                       // interval index, 0..1022
  if (m > NSEG - 1) m = NSEG - 1;
  float u   = (float)(sd - (double)m);                   // local coord in [0,1)
  float omu = 1.0f - u;
  float u2  = u * u;
  float u3  = u2 * u;
  const float S = 0.16666666666666666f;                  // 1/6
  float b[4];
  b[0] = omu * omu * omu * S;                            // B_{m-3}
  b[1] = fmaf(3.0f, u3, fmaf(-6.0f, u2, 4.0f)) * S;      // B_{m-2}
  b[2] = fmaf(-3.0f, u3, fmaf(3.0f, u2, fmaf(3.0f, u, 1.0f))) * S; // B_{m-1}
  b[3] = u3 * S;                                         // B_{m}
  float acc = 0.0f;
#pragma unroll
  for (int r = 0; r < 4; ++r) {
    int j = m - 3 + r;
    if (j >= 0 && j < NW) acc = fmaf(lw[j], b[r], acc);  // drop edge columns
  }
  return acc;
}

__global__ __launch_bounds__(BLOCK)
void bspline_act_kernel(const float* __restrict__ x,
                        const float* __restrict__ w,
                        float* __restrict__ out, int n) {
  __shared__ float lw[NW];
  stage_weights(w, lw);

  const int tid    = blockIdx.x * blockDim.x + threadIdx.x;
  const int stride = gridDim.x * blockDim.x;
  const int n4     = n >> 2;

  const float4* __restrict__ x4 = (const float4*)x;
  float4* __restrict__ o4       = (float4*)out;

  for (int i = tid; i < n4; i += stride) {            // coalesced 16B loads/stores
    float4 xv = x4[i];
    float4 ov;
    ov.x = bspline_eval(xv.x, lw);
    ov.y = bspline_eval(xv.y, lw);
    ov.z = bspline_eval(xv.z, lw);
    ov.w = bspline_eval(xv.w, lw);
    o4[i] = ov;
  }
  for (int i = (n4 << 2) + tid; i < n; i += stride)   // tail (n % 4)
    out[i] = bspline_eval(x[i], lw);
}

extern "C" void kernel_launch(void* const* d_in, const int* in_sizes, int n_in,
                              void* d_out, int out_size, void* d_ws, size_t ws_size,
                              hipStream_t stream) {
  const float* x = (const float*)d_in[0];   // [N,1] float32
  const float* w = (const float*)d_in[1];   // [1020] float32
  float* out     = (float*)d_out;           // [N,1] float32
  const int n    = out_size;                // N = 262144

  int n4   = n >> 2;
  int grid = (n4 + BLOCK - 1) / BLOCK;      // 256 blocks for N=262144
  if (grid < 1)    grid = 1;
  if (grid > 2048) grid = 2048;             // grid-stride covers the rest

  bspline_act_kernel<<<grid, BLOCK, 0, stream>>>(x, w, out, n);
}